// SamplePDF_9105330667610
// MI455X (gfx1250) — compile-verified
//
#include <hip/hip_runtime.h>

#define NCOARSE 64
#define NSAMP   128
#define NOUT    192
#define WAVES_PER_BLOCK 8
#define BLOCK_THREADS (WAVES_PER_BLOCK * 32)
#define RAYS_PER_WAVE 4
#define PREFETCH_RAYS 1024

// Wave-local LDS fence: same-wave LDS ops are processed in order on CDNA5;
// we need only a DScnt drain + compiler scheduling barrier (no s_barrier).
__device__ __forceinline__ void wave_lds_fence() {
    __builtin_amdgcn_wave_barrier();
    asm volatile("s_wait_dscnt 0x0" ::: "memory");
    __builtin_amdgcn_wave_barrier();
}

// u-grid value. fl(127 * fl(1/127)) == 1.0f exactly, so no endpoint case.
__device__ __forceinline__ float u_of(int s) {
    return (float)s * (1.0f / 127.0f);
}

// Exact smallest s in [0,128] with u_of(s) >= c (guess may be off by <=2).
__device__ __forceinline__ int smin_of(float c) {
    int g = (int)ceilf(c * 127.0f);
    g = (g < 0) ? 0 : ((g > NSAMP) ? NSAMP : g);
    #pragma unroll
    for (int r = 0; r < 2; ++r) g -= ((g > 0) && (u_of(g - 1) >= c)) ? 1 : 0;
    #pragma unroll
    for (int r = 0; r < 2; ++r) g += ((g < NSAMP) && (u_of(g) < c)) ? 1 : 0;
    return g;
}

__global__ __launch_bounds__(BLOCK_THREADS)
void nerf_sample_merge_kernel(const float* __restrict__ pi_g,   // [N,64] sorted depths
                              const float* __restrict__ w_g,    // [N,64] weights
                              float* __restrict__ out_g,        // [N,192]
                              int n_rays)
{
    // table entry b: (cdf[b], bins[b], pi[b], pi[b+1]); entry 63 is a sentinel:
    // (2.0, bins[62], pi[63], +INF) -> id==63 needs no special-casing anywhere.
    __shared__ __align__(16) float4 s_tb  [WAVES_PER_BLOCK][64];
    __shared__ __align__(16) int    s_cnt [WAVES_PER_BLOCK][NSAMP + 4]; // +dummy slots
    __shared__ __align__(16) int    s_cnt2[WAVES_PER_BLOCK][64];

    const int lane = threadIdx.x & 31;
    const int wv   = threadIdx.x >> 5;
    const int ray0 = (blockIdx.x * WAVES_PER_BLOCK + wv) * RAYS_PER_WAVE;
    const float INF = __int_as_float(0x7f800000);

    for (int r = 0; r < RAYS_PER_WAVE; ++r) {
        int ray = ray0 + r;
        if (ray >= n_rays) ray = n_rays - 1;   // tail: redo last ray (benign)

        // ---- coalesced input: 2 depths + 2 weights per lane ----
        const float2 p  = *(const float2*)(pi_g + (size_t)ray * NCOARSE + 2 * lane);
        const float2 ww = *(const float2*)(w_g  + (size_t)ray * NCOARSE + 2 * lane);

        {   // branchless L2 prime (gfx1250 global_prefetch_b8)
            int nr = ray + PREFETCH_RAYS;
            nr = (nr < n_rays) ? nr : (n_rays - 1);
            __builtin_prefetch(pi_g + (size_t)nr * NCOARSE + 2 * lane, 0, 0);
            __builtin_prefetch(w_g  + (size_t)nr * NCOARSE + 2 * lane, 0, 0);
        }

        // ---- zero counters (packed stores; dummy slots never read) ----
        *(int4*)&s_cnt [wv][4 * lane] = make_int4(0, 0, 0, 0);
        *(int2*)&s_cnt2[wv][2 * lane] = make_int2(0, 0);

        // ---- mids: bins[i] = 0.5*(pi[i]+pi[i+1]) ----
        float p_next = __shfl_down(p.x, 1, 32);
        float m0 = 0.5f * (p.x + p.y);             // bins[2l] (incl. l=31 -> bins[62])
        float m1 = 0.5f * (p.y + p_next);          // bins[2l+1] (invalid at l=31)

        // ---- pdf entries j=0..61 use w[j+1]+1e-5; lane l<31 owns j=2l, 2l+1 ----
        float w_next = __shfl_down(ww.x, 1, 32);
        float v0 = (lane < 31) ? (ww.y   + 1e-5f) : 0.0f;
        float v1 = (lane < 31) ? (w_next + 1e-5f) : 0.0f;

        // ---- wave32 inclusive scan of pair sums ----
        float pairsum = v0 + v1;
        float incl = pairsum;
        #pragma unroll
        for (int off = 1; off < 32; off <<= 1) {
            float t = __shfl_up(incl, off, 32);
            if (lane >= off) incl += t;
        }
        float total = __int_as_float(__builtin_amdgcn_readlane(__float_as_int(incl), 31));
        float excl  = incl - pairsum;

        float invT = 1.0f / total;
        float c0 = excl * invT;                    // cdf[2l]  (lane31 -> cdf[62] ~ 1)
        float c1 = (excl + v0) * invT;             // cdf[2l+1]
        float c1s = (lane < 31) ? c1 : 2.0f;       // entry63.x: cdf sentinel
        float m1s = (lane < 31) ? m1 : m0;         // entry63.y: bins[62] (zero-width lerp)
        float pn  = (lane < 31) ? p_next : INF;    // entry63.w: +INF (kills dup-term)

        // ---- table: two b128 per lane ----
        s_tb[wv][2 * lane]     = make_float4(c0,  m0,  p.x, p.y);
        s_tb[wv][2 * lane + 1] = make_float4(c1s, m1s, p.y, pn);

        // ---- exact u-grid thresholds; histogram (dummy-redirect, no exec ops) ----
        int g0 = smin_of(c0);                      // lane0: c0=0 -> 0
        int g1 = smin_of(c1s);                     // lane31: c=2 -> 128
        int h0 = (lane > 0  && g0 < NSAMP) ? g0 : NSAMP;      // b=2l   in [2,62]
        int h1 = (lane < 31 && g1 < NSAMP) ? g1 : NSAMP + 1;  // b=2l+1 in [1,61]
        atomicAdd(&s_cnt[wv][h0], 1);
        atomicAdd(&s_cnt[wv][h1], 1);

        wave_lds_fence();

        // ---- bin index for all 128 samples: in-lane prefix + wave32 scan ----
        int4 cv = *(const int4*)&s_cnt[wv][4 * lane];
        int q0 = cv.x, q1 = q0 + cv.y, q2 = q1 + cv.z, q3 = q2 + cv.w;
        int iscan = q3;
        #pragma unroll
        for (int off = 1; off < 32; off <<= 1) {
            int t = __shfl_up(iscan, off, 32);
            if (lane >= off) iscan += t;
        }
        int exclc = iscan - q3;
        int inds[4] = { 1 + exclc + q0, 1 + exclc + q1,
                        1 + exclc + q2, 1 + exclc + q3 };   // each in [1,63]

        const float4* tbw = s_tb[wv];
        float* orow = out_g + (size_t)ray * NOUT;

        // ---- samples s = 4*lane+k: interpolate, exact depth-rank, scatter ----
        #pragma unroll
        for (int k = 0; k < 4; ++k) {
            int s  = 4 * lane + k;
            int id = inds[k];
            float u = u_of(s);
            float2 pb = *(const float2*)&tbw[id - 1];  // (cdf_b, bins_b)
            float4 pa = tbw[id];                       // (cdf_a, bins_a, pi[id], pi[id+1])
            float denom = pa.x - pb.x;
            denom = (denom < 1e-5f) ? 1.0f : denom;
            float t   = (u - pb.x) * __builtin_amdgcn_rcpf(denom);
            float smp = pb.y + t * (pa.y - pb.y);

            // exact rank among the 64 depths: id + (pi[id] <= smp) (+ dup fix)
            float a1 = pa.z;                           // pi[id]
            float a2 = pa.w;                           // pi[id+1] or +INF at id==63
            int rank = id + ((a1 <= smp) ? 1 : 0) + ((a2 <= smp) ? 1 : 0);
            orow[s + rank] = smp;

            if (smp < a1) atomicAdd(&s_cnt2[wv][id], 1);   // feeds depth ranks
        }
        wave_lds_fence();

        // ---- depths: rank = smin[i-1] + cnt2[i] (register/shuffle bases) ----
        int g1_prev = __shfl_up(g1, 1, 32);        // smin[2l-1] from lane l-1
        int base0 = (lane == 0) ? 0 : g1_prev;     // i = 2l   -> smin[i-1]
        int base1 = g0;                            // i = 2l+1 -> smin[2l]
        int2 c2p = *(const int2*)&s_cnt2[wv][2 * lane];
        orow[2 * lane     + base0 + c2p.x] = p.x;
        orow[2 * lane + 1 + base1 + c2p.y] = p.y;

        wave_lds_fence();   // protect counter re-zeroing in next iteration
    }
}

extern "C" void kernel_launch(void* const* d_in, const int* in_sizes, int n_in,
                              void* d_out, int out_size, void* d_ws, size_t ws_size,
                              hipStream_t stream) {
    const float* pi = (const float*)d_in[0];   // point_interval [N,64] f32
    const float* w  = (const float*)d_in[1];   // weights        [N,64] f32
    // d_in[2] = perturb scalar (0) -> deterministic branch only
    float* out = (float*)d_out;                // [N,192] f32

    int n_rays = in_sizes[0] / NCOARSE;
    int rays_per_block = WAVES_PER_BLOCK * RAYS_PER_WAVE;
    int blocks = (n_rays + rays_per_block - 1) / rays_per_block;
    nerf_sample_merge_kernel<<<blocks, BLOCK_THREADS, 0, stream>>>(pi, w, out, n_rays);
}